// DGCNNControlPoints_4088808866085
// MI455X (gfx1250) — compile-verified
//
#include <hip/hip_runtime.h>
#include <math.h>

// ---------------------------------------------------------------------------
// DGCNN forward for MI455X (gfx1250, wave32).
// All dense GEMMs (pairwise-distance X·X^T, edge-conv 1x1 convs, 512->1024
// pointwise conv) run on V_WMMA_F32_16X16X4_F32 (fp32 matrix pipe: preserves
// exact KNN top-k selection and BN statistics).
// BN(gamma=1)+LeakyReLU is a monotone per-channel affine, so max-over-k is
// hoisted BEFORE the normalization: one GEMM pass produces both the running
// max and the per-channel sum/sum-of-squares needed for batch statistics.
// Workspace requirement: ~61 MB. Fully deterministic (no float atomics).
// ---------------------------------------------------------------------------

typedef float v2f __attribute__((ext_vector_type(2)));
typedef float v8f __attribute__((ext_vector_type(8)));

#define NPTS   2048
#define BATCH  4
#define KNN    40
#define BN_TOT (BATCH * NPTS)   // 8192

__device__ __forceinline__ v8f wmma_f32(v2f a, v2f b, v8f c) {
  // D = A(16x4) * B(4x16) + C(16x16), fp32, wave32.
  return __builtin_amdgcn_wmma_f32_16x16x4_f32(false, a, false, b, (short)0, c,
                                               false, false);
}

// ---------------------------------------------------------------------------
// x (B,3,N) -> xt0 (B,N,3)
__global__ void transpose3_kernel(const float* __restrict__ x,
                                  float* __restrict__ xt, int total) {
  int i = blockIdx.x * 256 + threadIdx.x;
  if (i >= total) return;
  int c = i % 3, t = i / 3;
  int n = t % NPTS, b = t / NPTS;
  xt[i] = x[((size_t)b * 3 + c) * NPTS + n];
}

// sq[b,n] = sum_c xt[b,n,c]^2
__global__ void sqnorm_kernel(const float* __restrict__ xt,
                              float* __restrict__ sq, int CIN, int total) {
  int i = blockIdx.x * 256 + threadIdx.x;
  if (i >= total) return;
  const float* r = xt + (size_t)i * CIN;
  float s = 0.f;
  for (int c = 0; c < CIN; ++c) s = fmaf(r[c], r[c], s);
  sq[i] = s;
}

// ---------------------------------------------------------------------------
// Pairwise "distance" pd[n,m] = 2*<x_n,x_m> - |x_n|^2 - |x_m|^2, one batch b.
// Small-C (=3) scalar version for layer 1.
__global__ void dist_small_kernel(const float* __restrict__ xt,
                                  const float* __restrict__ sq,
                                  float* __restrict__ pd, int b) {
  size_t i = (size_t)blockIdx.x * 256 + threadIdx.x;
  if (i >= (size_t)NPTS * NPTS) return;
  int m = (int)(i % NPTS);
  int n = (int)(i / NPTS);
  const float* a = xt + ((size_t)b * NPTS + n) * 3;
  const float* c = xt + ((size_t)b * NPTS + m) * 3;
  float inner = a[0] * c[0] + a[1] * c[1] + a[2] * c[2];
  pd[i] = 2.f * inner - sq[b * NPTS + n] - sq[b * NPTS + m];
}

// WMMA version: S = Xb * Xb^T for one batch, fused pd epilogue.
// grid = (NPTS/16/8, NPTS/16), block = 256 (8 waves, one 16x16 tile each).
template <int CIN>
__global__ __launch_bounds__(256) void dist_wmma_kernel(
    const float* __restrict__ xt, const float* __restrict__ sq,
    float* __restrict__ pd, int b) {
  const int lane = threadIdx.x & 31;
  const int wave = threadIdx.x >> 5;
  const int mt = blockIdx.x * 8 + wave;  // row tile
  const int nt = blockIdx.y;             // col tile
  const int laneN = lane & 15;
  const int koff = (lane < 16) ? 0 : 2;
  const int moff = (lane < 16) ? 0 : 8;

  const float* Arow = xt + ((size_t)(b * NPTS + mt * 16 + laneN)) * CIN;
  const float* Brow = xt + ((size_t)(b * NPTS + nt * 16 + laneN)) * CIN;

  v8f acc = {};
  for (int k0 = 0; k0 < CIN; k0 += 4) {
    float2 af = *reinterpret_cast<const float2*>(Arow + k0 + koff);
    float2 bf = *reinterpret_cast<const float2*>(Brow + k0 + koff);
    v2f a; a[0] = af.x; a[1] = af.y;
    v2f w; w[0] = bf.x; w[1] = bf.y;
    acc = wmma_f32(a, w, acc);
  }
  const int mcol = nt * 16 + laneN;
  const float sqm = sq[b * NPTS + mcol];
#pragma unroll
  for (int j = 0; j < 8; ++j) {
    int nrow = mt * 16 + j + moff;
    pd[(size_t)nrow * NPTS + mcol] = 2.f * acc[j] - sq[b * NPTS + nrow] - sqm;
  }
}

// ---------------------------------------------------------------------------
// Top-40 per row (iterative argmax, lowest-index tie-break, matches top_k).
// One block per point; row lives in LDS.
__global__ __launch_bounds__(256) void topk_kernel(const float* __restrict__ pd,
                                                   int* __restrict__ idx,
                                                   int b) {
  __shared__ float vals[NPTS];
  __shared__ float rv[256];
  __shared__ int ri[256];
  const int n = blockIdx.x;
  const int tid = threadIdx.x;
  const float* row = pd + (size_t)n * NPTS;
  for (int i = tid; i < NPTS; i += 256) vals[i] = row[i];
  __syncthreads();
  for (int t = 0; t < KNN; ++t) {
    float bv = -INFINITY;
    int bi = NPTS;
    for (int i = tid; i < NPTS; i += 256) {
      float v = vals[i];
      if (v > bv) { bv = v; bi = i; }
    }
    rv[tid] = bv; ri[tid] = bi;
    __syncthreads();
    for (int s = 128; s > 0; s >>= 1) {
      if (tid < s) {
        float ov = rv[tid + s]; int oi = ri[tid + s];
        if (ov > rv[tid] || (ov == rv[tid] && oi < ri[tid])) {
          rv[tid] = ov; ri[tid] = oi;
        }
      }
      __syncthreads();
    }
    if (tid == 0) {
      idx[((size_t)b * NPTS + n) * KNN + t] = ri[0];
      vals[ri[0]] = -INFINITY;
    }
    __syncthreads();
  }
}

// ---------------------------------------------------------------------------
// EdgeConv: one block per (b,n). Edge features (neighbor-center | center)
// staged in LDS; each wave owns full 16-channel column strips (3 row tiles of
// the 40->48 padded k dimension share each weight load). Produces per-(b,n,o)
// max over k plus per-channel sum / sumsq partials (k>=40 rows masked).
template <int CIN, int COUT>
__global__ __launch_bounds__(256) void edgeconv_kernel(
    const float* __restrict__ xt, const int* __restrict__ idx,
    const float* __restrict__ W, float* __restrict__ maxbuf,
    float* __restrict__ sumbuf, float* __restrict__ sqbuf) {
  constexpr int KF = ((2 * CIN + 7) / 8) * 8;  // padded K (layer1: 6 -> 8)
  __shared__ float s_nm[48][CIN];  // neighbor - center, zero for kk >= 40
  __shared__ float s_ct[CIN];      // center features
  __shared__ int s_idx[KNN];

  const int bn = blockIdx.x;
  const int b = bn >> 11;  // N = 2048
  const int tid = threadIdx.x;

  for (int c = tid; c < CIN; c += 256) s_ct[c] = xt[(size_t)bn * CIN + c];
  if (tid < KNN) s_idx[tid] = idx[(size_t)bn * KNN + tid];
  __syncthreads();
  for (int e = tid; e < 48 * CIN; e += 256) {
    int kk = e / CIN, c = e - kk * CIN;
    float v = 0.f;
    if (kk < KNN) {
      int j = s_idx[kk];
      v = xt[((size_t)b * NPTS + j) * CIN + c] - s_ct[c];
    }
    s_nm[kk][c] = v;
  }
  __syncthreads();

  const int lane = tid & 31;
  const int wave = tid >> 5;
  const int laneN = lane & 15;
  const int koff = (lane < 16) ? 0 : 2;
  const int moff = (lane < 16) ? 0 : 8;

  for (int nt = wave; nt < COUT / 16; nt += 8) {
    const int o = nt * 16 + laneN;
    v8f acc0 = {}, acc1 = {}, acc2 = {};
    for (int k0 = 0; k0 < KF; k0 += 4) {
      const int ka = k0 + koff;
      v2f bv;
      bv[0] = (ka < 2 * CIN) ? W[(size_t)o * (2 * CIN) + ka] : 0.f;
      bv[1] = (ka + 1 < 2 * CIN) ? W[(size_t)o * (2 * CIN) + ka + 1] : 0.f;
      auto lda = [&](int m) -> v2f {
        v2f r;
        int c0 = ka, c1 = ka + 1;
        r[0] = (c0 < CIN) ? s_nm[m][c0]
                          : ((c0 < 2 * CIN) ? s_ct[c0 - CIN] : 0.f);
        r[1] = (c1 < CIN) ? s_nm[m][c1]
                          : ((c1 < 2 * CIN) ? s_ct[c1 - CIN] : 0.f);
        return r;
      };
      acc0 = wmma_f32(lda(0 * 16 + laneN), bv, acc0);
      acc1 = wmma_f32(lda(1 * 16 + laneN), bv, acc1);
      acc2 = wmma_f32(lda(2 * 16 + laneN), bv, acc2);
    }
    // Per-lane stats over valid kk (<40), then fold lanes L / L+16 (same o).
    float s = 0.f, q = 0.f, mx = -INFINITY;
    v8f accs[3] = {acc0, acc1, acc2};
#pragma unroll
    for (int mtile = 0; mtile < 3; ++mtile) {
#pragma unroll
      for (int j = 0; j < 8; ++j) {
        int kk = mtile * 16 + j + moff;
        if (kk < KNN) {
          float v = accs[mtile][j];
          s += v; q += v * v; mx = fmaxf(mx, v);
        }
      }
    }
    s += __shfl_xor(s, 16, 32);
    q += __shfl_xor(q, 16, 32);
    mx = fmaxf(mx, __shfl_xor(mx, 16, 32));
    if (lane < 16) {
      size_t off = (size_t)bn * COUT + o;
      maxbuf[off] = mx; sumbuf[off] = s; sqbuf[off] = q;
    }
  }
}

// ---------------------------------------------------------------------------
// Conv5: h = W5(1024x512) * concat(x1,x2,x3,x4); fused global-max over N plus
// per-(b,channel) BN partials. grid = (64 channel tiles, 4 batches).
__global__ __launch_bounds__(256) void conv5_kernel(
    const float* __restrict__ x1, const float* __restrict__ x2,
    const float* __restrict__ x3, const float* __restrict__ x4,
    const float* __restrict__ W5, float* __restrict__ maxbuf,
    float* __restrict__ sumbuf, float* __restrict__ sqbuf) {
  const int nt = blockIdx.x;
  const int b = blockIdx.y;
  const int tid = threadIdx.x;
  const int lane = tid & 31;
  const int wave = tid >> 5;
  const int laneN = lane & 15;
  const int koff = (lane < 16) ? 0 : 2;
  const int o = nt * 16 + laneN;

  float s = 0.f, q = 0.f, mx = -INFINITY;
  for (int mt = wave; mt < NPTS / 16; mt += 8) {
    const int nrowA = mt * 16 + laneN;
    v8f acc = {};
    auto kblock = [&](const float* src, int CL, int cbase, int kcnt) {
      const float* arow = src + ((size_t)(b * NPTS + nrowA)) * CL;
      const float* wrow = W5 + (size_t)o * 512 + cbase;
      for (int c = 0; c < kcnt; c += 4) {
        float2 af = *reinterpret_cast<const float2*>(arow + c + koff);
        float2 wf = *reinterpret_cast<const float2*>(wrow + c + koff);
        v2f a; a[0] = af.x; a[1] = af.y;
        v2f w; w[0] = wf.x; w[1] = wf.y;
        acc = wmma_f32(a, w, acc);
      }
    };
    kblock(x1, 64, 0, 64);
    kblock(x2, 64, 64, 64);
    kblock(x3, 128, 128, 128);
    kblock(x4, 256, 256, 256);
#pragma unroll
    for (int j = 0; j < 8; ++j) {
      float v = acc[j];
      s += v; q += v * v; mx = fmaxf(mx, v);
    }
  }
  s += __shfl_xor(s, 16, 32);
  q += __shfl_xor(q, 16, 32);
  mx = fmaxf(mx, __shfl_xor(mx, 16, 32));

  __shared__ float rs[8][16], rq[8][16], rm[8][16];
  if (lane < 16) { rs[wave][laneN] = s; rq[wave][laneN] = q; rm[wave][laneN] = mx; }
  __syncthreads();
  if (tid < 16) {
    float S = 0.f, Q = 0.f, M = -INFINITY;
    for (int w = 0; w < 8; ++w) {
      S += rs[w][tid]; Q += rq[w][tid]; M = fmaxf(M, rm[w][tid]);
    }
    size_t off = (size_t)b * 1024 + nt * 16 + tid;
    maxbuf[off] = M; sumbuf[off] = S; sqbuf[off] = Q;
  }
}

// ---------------------------------------------------------------------------
// Reduce BN partials (deterministic fixed-order tree) -> scale/shift.
__global__ __launch_bounds__(256) void bnstats_kernel(
    const float* __restrict__ sumbuf, const float* __restrict__ sqbuf,
    const float* __restrict__ gamma, const float* __restrict__ beta,
    float* __restrict__ scsh, int nrows, int COUT, float invR) {
  const int o = blockIdx.x;
  float s = 0.f, q = 0.f;
  for (int i = threadIdx.x; i < nrows; i += 256) {
    s += sumbuf[(size_t)i * COUT + o];
    q += sqbuf[(size_t)i * COUT + o];
  }
  __shared__ float ls[256], lq[256];
  ls[threadIdx.x] = s; lq[threadIdx.x] = q;
  __syncthreads();
  for (int st = 128; st > 0; st >>= 1) {
    if (threadIdx.x < st) {
      ls[threadIdx.x] += ls[threadIdx.x + st];
      lq[threadIdx.x] += lq[threadIdx.x + st];
    }
    __syncthreads();
  }
  if (threadIdx.x == 0) {
    float mean = ls[0] * invR;
    float var = lq[0] * invR - mean * mean;
    float sc = gamma[o] * rsqrtf(var + 1e-5f);
    scsh[2 * o] = sc;
    scsh[2 * o + 1] = beta[o] - mean * sc;
  }
}

// out = lrelu(scale*max + shift)   (monotone affine hoisted past max)
__global__ void bnapply_lrelu_kernel(const float* __restrict__ maxbuf,
                                     const float* __restrict__ scsh,
                                     float* __restrict__ out, int total,
                                     int COUT) {
  int i = blockIdx.x * 256 + threadIdx.x;
  if (i >= total) return;
  int o = i % COUT;
  float v = scsh[2 * o] * maxbuf[i] + scsh[2 * o + 1];
  out[i] = (v >= 0.f) ? v : 0.2f * v;
}

// ---------------------------------------------------------------------------
// Head: raw[b,o] = W(O,C) * in[b,:] + bias
__global__ void gemv_kernel(const float* __restrict__ W,
                            const float* __restrict__ bias,
                            const float* __restrict__ in,
                            float* __restrict__ out, int O, int C) {
  int i = blockIdx.x * 256 + threadIdx.x;
  if (i >= BATCH * O) return;
  int b = i / O, o = i % O;
  const float* wr = W + (size_t)o * C;
  const float* xr = in + (size_t)b * C;
  float s = bias[o];
  for (int c = 0; c < C; ++c) s = fmaf(wr[c], xr[c], s);
  out[i] = s;
}

// BN over the batch of 4 + ReLU.
__global__ void headbn_relu_kernel(const float* __restrict__ raw,
                                   const float* __restrict__ gamma,
                                   const float* __restrict__ beta,
                                   float* __restrict__ out, int O) {
  int o = blockIdx.x * 256 + threadIdx.x;
  if (o >= O) return;
  float v0 = raw[o], v1 = raw[O + o], v2 = raw[2 * O + o], v3 = raw[3 * O + o];
  float m = 0.25f * (v0 + v1 + v2 + v3);
  float var = 0.25f * (v0 * v0 + v1 * v1 + v2 * v2 + v3 * v3) - m * m;
  float sc = gamma[o] * rsqrtf(var + 1e-5f);
  float sh = beta[o] - m * sc;
  out[o] = fmaxf(sc * v0 + sh, 0.f);
  out[O + o] = fmaxf(sc * v1 + sh, 0.f);
  out[2 * O + o] = fmaxf(sc * v2 + sh, 0.f);
  out[3 * O + o] = fmaxf(sc * v3 + sh, 0.f);
}

__global__ void final_kernel(const float* __restrict__ W8,
                             const float* __restrict__ b8,
                             const float* __restrict__ h7,
                             float* __restrict__ out) {
  int i = blockIdx.x * 256 + threadIdx.x;
  if (i >= BATCH * 1200) return;
  int b = i / 1200, j = i % 1200;
  const float* wr = W8 + (size_t)j * 1024;
  const float* xr = h7 + (size_t)b * 1024;
  float s = b8[j];
  for (int c = 0; c < 1024; ++c) s = fmaf(wr[c], xr[c], s);
  out[i] = tanhf(s);
}

// ---------------------------------------------------------------------------
extern "C" void kernel_launch(void* const* d_in, const int* in_sizes, int n_in,
                              void* d_out, int out_size, void* d_ws,
                              size_t ws_size, hipStream_t stream) {
  (void)in_sizes; (void)n_in; (void)out_size; (void)ws_size;
  const float* x  = (const float*)d_in[0];
  const float* w1 = (const float*)d_in[1];
  const float* g1 = (const float*)d_in[2];
  const float* b1 = (const float*)d_in[3];
  const float* w2 = (const float*)d_in[4];
  const float* g2 = (const float*)d_in[5];
  const float* b2 = (const float*)d_in[6];
  const float* w3 = (const float*)d_in[7];
  const float* g3 = (const float*)d_in[8];
  const float* b3 = (const float*)d_in[9];
  const float* w4 = (const float*)d_in[10];
  const float* g4 = (const float*)d_in[11];
  const float* b4 = (const float*)d_in[12];
  const float* w5 = (const float*)d_in[13];
  const float* g5 = (const float*)d_in[14];
  const float* b5 = (const float*)d_in[15];
  const float* w6 = (const float*)d_in[16];
  const float* bias6 = (const float*)d_in[17];
  const float* g6 = (const float*)d_in[18];
  const float* b6 = (const float*)d_in[19];
  const float* w7 = (const float*)d_in[20];
  const float* bias7 = (const float*)d_in[21];
  const float* g7 = (const float*)d_in[22];
  const float* b7 = (const float*)d_in[23];
  const float* w8 = (const float*)d_in[24];
  const float* bias8 = (const float*)d_in[25];

  // Workspace carve-out (~61 MB total).
  char* ws = (char*)d_ws;
  size_t off = 0;
  auto carve = [&](size_t bytes) -> void* {
    void* p = ws + off;
    off += (bytes + 255) & ~(size_t)255;
    return p;
  };
  float* pd   = (float*)carve((size_t)NPTS * NPTS * 4);        // 16.8 MB (one batch)
  float* sq   = (float*)carve((size_t)BN_TOT * 4);
  int*   kidx = (int*)  carve((size_t)BN_TOT * KNN * 4);       // 1.3 MB
  float* xt0  = (float*)carve((size_t)BN_TOT * 3 * 4);
  float* x1b  = (float*)carve((size_t)BN_TOT * 64 * 4);        // 2 MB
  float* x2b  = (float*)carve((size_t)BN_TOT * 64 * 4);        // 2 MB
  float* x3b  = (float*)carve((size_t)BN_TOT * 128 * 4);       // 4 MB
  float* x4b  = (float*)carve((size_t)BN_TOT * 256 * 4);       // 8 MB
  float* maxb = (float*)carve((size_t)BN_TOT * 256 * 4);       // 8 MB (reused)
  float* sumb = (float*)carve((size_t)BN_TOT * 256 * 4);       // 8 MB
  float* sqb  = (float*)carve((size_t)BN_TOT * 256 * 4);       // 8 MB
  float* scsh = (float*)carve(2 * 1024 * 4);
  float* g5v  = (float*)carve(BATCH * 1024 * 4);
  float* raw6 = (float*)carve(BATCH * 1024 * 4);
  float* h6   = (float*)carve(BATCH * 1024 * 4);
  float* raw7 = (float*)carve(BATCH * 1024 * 4);
  float* h7   = (float*)carve(BATCH * 1024 * 4);

  const float invR_edge = 1.f / (float)(BN_TOT * KNN);

  transpose3_kernel<<<(BN_TOT * 3 + 255) / 256, 256, 0, stream>>>(x, xt0, BN_TOT * 3);

  // ---- EdgeConv 1 (CIN=3 -> 64) ----
  sqnorm_kernel<<<(BN_TOT + 255) / 256, 256, 0, stream>>>(xt0, sq, 3, BN_TOT);
  for (int b = 0; b < BATCH; ++b) {
    dist_small_kernel<<<((size_t)NPTS * NPTS + 255) / 256, 256, 0, stream>>>(xt0, sq, pd, b);
    topk_kernel<<<NPTS, 256, 0, stream>>>(pd, kidx, b);
  }
  edgeconv_kernel<3, 64><<<BN_TOT, 256, 0, stream>>>(xt0, kidx, w1, maxb, sumb, sqb);
  bnstats_kernel<<<64, 256, 0, stream>>>(sumb, sqb, g1, b1, scsh, BN_TOT, 64, invR_edge);
  bnapply_lrelu_kernel<<<(BN_TOT * 64 + 255) / 256, 256, 0, stream>>>(maxb, scsh, x1b, BN_TOT * 64, 64);

  // ---- EdgeConv 2 (64 -> 64) ----
  sqnorm_kernel<<<(BN_TOT + 255) / 256, 256, 0, stream>>>(x1b, sq, 64, BN_TOT);
  for (int b = 0; b < BATCH; ++b) {
    dist_wmma_kernel<64><<<dim3(16, 128), 256, 0, stream>>>(x1b, sq, pd, b);
    topk_kernel<<<NPTS, 256, 0, stream>>>(pd, kidx, b);
  }
  edgeconv_kernel<64, 64><<<BN_TOT, 256, 0, stream>>>(x1b, kidx, w2, maxb, sumb, sqb);
  bnstats_kernel<<<64, 256, 0, stream>>>(sumb, sqb, g2, b2, scsh, BN_TOT, 64, invR_edge);
  bnapply_lrelu_kernel<<<(BN_TOT * 64 + 255) / 256, 256, 0, stream>>>(maxb, scsh, x2b, BN_TOT * 64, 64);

  // ---- EdgeConv 3 (64 -> 128) ----
  sqnorm_kernel<<<(BN_TOT + 255) / 256, 256, 0, stream>>>(x2b, sq, 64, BN_TOT);
  for (int b = 0; b < BATCH; ++b) {
    dist_wmma_kernel<64><<<dim3(16, 128), 256, 0, stream>>>(x2b, sq, pd, b);
    topk_kernel<<<NPTS, 256, 0, stream>>>(pd, kidx, b);
  }
  edgeconv_kernel<64, 128><<<BN_TOT, 256, 0, stream>>>(x2b, kidx, w3, maxb, sumb, sqb);
  bnstats_kernel<<<128, 256, 0, stream>>>(sumb, sqb, g3, b3, scsh, BN_TOT, 128, invR_edge);
  bnapply_lrelu_kernel<<<(BN_TOT * 128 + 255) / 256, 256, 0, stream>>>(maxb, scsh, x3b, BN_TOT * 128, 128);

  // ---- EdgeConv 4 (128 -> 256) ----
  sqnorm_kernel<<<(BN_TOT + 255) / 256, 256, 0, stream>>>(x3b, sq, 128, BN_TOT);
  for (int b = 0; b < BATCH; ++b) {
    dist_wmma_kernel<128><<<dim3(16, 128), 256, 0, stream>>>(x3b, sq, pd, b);
    topk_kernel<<<NPTS, 256, 0, stream>>>(pd, kidx, b);
  }
  edgeconv_kernel<128, 256><<<BN_TOT, 256, 0, stream>>>(x3b, kidx, w4, maxb, sumb, sqb);
  bnstats_kernel<<<256, 256, 0, stream>>>(sumb, sqb, g4, b4, scsh, BN_TOT, 256, invR_edge);
  bnapply_lrelu_kernel<<<(BN_TOT * 256 + 255) / 256, 256, 0, stream>>>(maxb, scsh, x4b, BN_TOT * 256, 256);

  // ---- Conv5 (concat 512 -> 1024) + global max over N ----
  conv5_kernel<<<dim3(64, BATCH), 256, 0, stream>>>(x1b, x2b, x3b, x4b, w5, maxb, sumb, sqb);
  bnstats_kernel<<<1024, 256, 0, stream>>>(sumb, sqb, g5, b5, scsh, BATCH, 1024, 1.f / (float)BN_TOT);
  bnapply_lrelu_kernel<<<(BATCH * 1024 + 255) / 256, 256, 0, stream>>>(maxb, scsh, g5v, BATCH * 1024, 1024);

  // ---- MLP head ----
  gemv_kernel<<<(BATCH * 1024 + 255) / 256, 256, 0, stream>>>(w6, bias6, g5v, raw6, 1024, 1024);
  headbn_relu_kernel<<<(1024 + 255) / 256, 256, 0, stream>>>(raw6, g6, b6, h6, 1024);
  gemv_kernel<<<(BATCH * 1024 + 255) / 256, 256, 0, stream>>>(w7, bias7, h6, raw7, 1024, 1024);
  headbn_relu_kernel<<<(1024 + 255) / 256, 256, 0, stream>>>(raw7, g7, b7, h7, 1024);
  final_kernel<<<(BATCH * 1200 + 255) / 256, 256, 0, stream>>>(w8, bias8, h7, (float*)d_out);
}